// MHAttentionMap_28664611733909
// MI455X (gfx1250) — compile-verified
//
#include <hip/hip_runtime.h>
#include <hip/hip_bf16.h>
#include <math.h>

// ---------------------------------------------------------------------------
// MHAttentionMap for MI455X (gfx1250, wave32).
// f32 WMMA (V_WMMA_F32_16X16X4_F32) for all three GEMMs -- the kernel is
// bandwidth-bound on the 384MB softmax output, so f32 matrix math is free
// and stays faithful to the f32 reference. GEMMs are M-blocked x4 per wave
// so each (expensive, strided) B fragment load feeds 4 WMMAs.
// ---------------------------------------------------------------------------

typedef __attribute__((ext_vector_type(2))) float v2f;
typedef __attribute__((ext_vector_type(8))) float v8f;

#define B_    4
#define Q_    300
#define QPAD_ 320          // 20 tiles of 16 -> 5 blocks of 64 for scores
#define CDIM_ 256
#define NH_   8
#define HD_   32
#define HW_   10000        // 100*100 = 625 tiles of 16
#define NORM_ 0.17677669529663687f   // (256/8)^-0.5

#define WMMA_F32(a, b, c) \
  __builtin_amdgcn_wmma_f32_16x16x4_f32(false, (a), false, (b), (short)0, (c), false, false)

// ---------------------------------------------------------------------------
// K1: qp[b][q][h] = (sum_c q[b][q][c] * q_w[h][c] + q_b[h]) * NORM
// Small GEMM (0.16 GFLOP): one wave per 16x16 tile. grid (20, 16, 4).
__global__ __launch_bounds__(32)
void qproj_kernel(const float* __restrict__ q, const float* __restrict__ qw,
                  const float* __restrict__ qb, float* __restrict__ qp) {
  const int lane = threadIdx.x;
  const int rA   = lane & 15;      // A row / B col / D col
  const int kHi  = lane >> 4;      // K sub-offset selector
  const int q0   = blockIdx.x * 16;
  const int n0   = blockIdx.y * 16;
  const int b    = blockIdx.z;

  const int qr = q0 + rA;
  const int qc = qr < Q_ ? qr : (Q_ - 1);          // clamp (no OOB), zero later
  const float valid = (qr < Q_) ? 1.0f : 0.0f;
  const float* __restrict__ aptr = q  + ((size_t)b * Q_ + qc) * CDIM_;
  const float* __restrict__ bptr = qw + (size_t)(n0 + rA) * CDIM_;

  v8f c = {};
  #pragma unroll 4
  for (int k0 = 0; k0 < CDIM_; k0 += 4) {
    const int kk = k0 + 2 * kHi;
    v2f a = *(const v2f*)(aptr + kk);
    a.x *= valid; a.y *= valid;                    // pad rows contribute zero
    v2f bf = *(const v2f*)(bptr + kk);
    c = WMMA_F32(a, bf, c);
  }

  const float bias = qb[n0 + rA];
  #pragma unroll
  for (int v = 0; v < 8; ++v) {
    const int qrow = q0 + v + 8 * kHi;             // <= 319 < QPAD_
    qp[((size_t)b * QPAD_ + qrow) * CDIM_ + n0 + rA] = (c[v] + bias) * NORM_;
  }
}

// ---------------------------------------------------------------------------
// K2: kp[b][o][hw] = sum_c k_w[o][c] * k[b][c][hw] + k_b[o]
// M-blocked x4: one wave computes a 64x16 slab; the strided B fragment
// (from k, 41MB) is loaded once per k-step and feeds 4 WMMAs.
// grid (4, 625, 4), block 32.
__global__ __launch_bounds__(32)
void kproj_kernel(const float* __restrict__ k, const float* __restrict__ kw,
                  const float* __restrict__ kb, float* __restrict__ kp) {
  const int lane = threadIdx.x;
  const int rA   = lane & 15;
  const int kHi  = lane >> 4;
  const int m0   = blockIdx.x * 64;  // output channel slab
  const int n0   = blockIdx.y * 16;  // hw tile
  const int b    = blockIdx.z;

  const float* __restrict__ abase = kw + (size_t)(m0 + rA) * CDIM_;
  const float* __restrict__ bbase = k + (size_t)b * CDIM_ * HW_ + (n0 + rA);

  v8f c[4] = {};
  #pragma unroll 2
  for (int k0 = 0; k0 < CDIM_; k0 += 4) {
    const int kk = k0 + 2 * kHi;
    v2f bf;
    bf.x = bbase[(size_t)kk * HW_];
    bf.y = bbase[(size_t)(kk + 1) * HW_];
    #pragma unroll
    for (int j = 0; j < 4; ++j) {
      v2f a = *(const v2f*)(abase + (size_t)j * 16 * CDIM_ + kk);
      c[j] = WMMA_F32(a, bf, c[j]);
    }
  }

  float* __restrict__ outp = kp + (size_t)b * CDIM_ * HW_ + n0 + rA;
  #pragma unroll
  for (int j = 0; j < 4; ++j) {
    #pragma unroll
    for (int v = 0; v < 8; ++v) {
      const int m = m0 + j * 16 + v + 8 * kHi;
      outp[(size_t)m * HW_] = c[j][v] + kb[m];
    }
  }
}

// ---------------------------------------------------------------------------
// K3: logits[b][q][h][hw] = sum_c qp[b][q][h*32+c] * kp[b][h*32+c][hw],
// masked -> -inf. M-blocked x4 (64 queries x 16 hw per wave): the strided
// kp fragment is loaded once per k-step for 4 WMMAs.
// grid (5, 625, 32=b*8), block 32.
__global__ __launch_bounds__(32)
void scores_kernel(const float* __restrict__ qp, const float* __restrict__ kp,
                   const unsigned char* __restrict__ mask, float* __restrict__ out) {
  const int lane = threadIdx.x;
  const int rA   = lane & 15;
  const int kHi  = lane >> 4;
  const int q0   = blockIdx.x * 64;
  const int n0   = blockIdx.y * 16;
  const int b    = blockIdx.z >> 3;
  const int h    = blockIdx.z & 7;

  const float* __restrict__ abase = qp + ((size_t)b * QPAD_ + q0 + rA) * CDIM_ + h * HD_;
  const float* __restrict__ bbase = kp + ((size_t)b * CDIM_ + h * HD_) * HW_ + (n0 + rA);

  v8f c[4] = {};
  #pragma unroll
  for (int k0 = 0; k0 < HD_; k0 += 4) {
    const int kk = k0 + 2 * kHi;
    v2f bf;
    bf.x = bbase[(size_t)kk * HW_];
    bf.y = bbase[(size_t)(kk + 1) * HW_];
    #pragma unroll
    for (int j = 0; j < 4; ++j) {
      v2f a = *(const v2f*)(abase + (size_t)j * 16 * CDIM_ + kk);
      c[j] = WMMA_F32(a, bf, c[j]);
    }
  }

  const int hw = n0 + rA;
  const bool msk = mask[(size_t)b * HW_ + hw] != 0;
  const float NEG_INF = -__builtin_inff();
  #pragma unroll
  for (int j = 0; j < 4; ++j) {
    #pragma unroll
    for (int v = 0; v < 8; ++v) {
      const int qq = q0 + j * 16 + v + 8 * kHi;
      if (qq < Q_) {
        out[(((size_t)b * Q_ + qq) * NH_ + h) * HW_ + hw] = msk ? NEG_INF : c[j][v];
      }
    }
  }
}

// ---------------------------------------------------------------------------
// K4: per (b,q) row of 80000 logits: online (max, sum-exp), float4 reads.
// grid 1200, block 256.
__global__ __launch_bounds__(256)
void rowstat_kernel(const float* __restrict__ out, float* __restrict__ stats) {
  const int row = blockIdx.x;                       // 0..1199
  const int tid = threadIdx.x;
  const float4* __restrict__ p4 = (const float4*)(out + (size_t)row * (NH_ * HW_));
  const float NEG_INF = -__builtin_inff();

  float m = NEG_INF, s = 0.0f;
  #define ONLINE_(X) { const float xx = (X);                                   \
      if (xx > m)             { s = s * __expf(m - xx) + 1.0f; m = xx; }       \
      else if (xx != NEG_INF) { s += __expf(xx - m); } }
  for (int i = tid; i < (NH_ * HW_) / 4; i += 256) {
    const float4 x = p4[i];
    ONLINE_(x.x); ONLINE_(x.y); ONLINE_(x.z); ONLINE_(x.w);
  }
  #undef ONLINE_

  __shared__ float sm[256];
  __shared__ float ss[256];
  sm[tid] = m; ss[tid] = s;
  __syncthreads();
  for (int off = 128; off > 0; off >>= 1) {
    if (tid < off) {
      const float m2 = sm[tid + off], s2 = ss[tid + off];
      const float M  = fmaxf(m, m2);
      const float sa = (s  > 0.0f) ? s  * __expf(m  - M) : 0.0f;
      const float sb = (s2 > 0.0f) ? s2 * __expf(m2 - M) : 0.0f;
      m = M; s = sa + sb;
      sm[tid] = m; ss[tid] = s;
    }
    __syncthreads();
  }
  if (tid == 0) { stats[2 * row] = m; stats[2 * row + 1] = s; }
}

// ---------------------------------------------------------------------------
// K5: out = exp(x - m) / s, float4 vectorized (80000 % 4 == 0 -> no row split).
// 96e6/4 = 24e6 float4s; grid 93750, block 256 (exact).
__global__ __launch_bounds__(256)
void norm_kernel(float* __restrict__ out, const float* __restrict__ stats) {
  const unsigned v = blockIdx.x * 256u + threadIdx.x;   // float4 index
  if (v >= 24000000u) return;
  const unsigned row = v / 20000u;                      // 80000/4 per row
  const float m   = stats[2 * row];
  const float inv = 1.0f / stats[2 * row + 1];
  float4* p = (float4*)out;
  float4 x = p[v];
  x.x = __expf(x.x - m) * inv;
  x.y = __expf(x.y - m) * inv;
  x.z = __expf(x.z - m) * inv;
  x.w = __expf(x.w - m) * inv;
  p[v] = x;
}

// ---------------------------------------------------------------------------
extern "C" void kernel_launch(void* const* d_in, const int* in_sizes, int n_in,
                              void* d_out, int out_size, void* d_ws, size_t ws_size,
                              hipStream_t stream) {
  (void)in_sizes; (void)n_in; (void)out_size; (void)ws_size;

  const float*         q    = (const float*)d_in[0];          // [4,300,256]
  const float*         k    = (const float*)d_in[1];          // [4,256,100,100]
  const unsigned char* mask = (const unsigned char*)d_in[2];  // [4,100,100] bool (1B)
  const float*         q_w  = (const float*)d_in[3];          // [256,256]
  const float*         q_b  = (const float*)d_in[4];          // [256]
  const float*         k_w  = (const float*)d_in[5];          // [256,256]
  const float*         k_b  = (const float*)d_in[6];          // [256]
  float*               out  = (float*)d_out;                  // [4,300,8,100,100]

  // Workspace layout (~42.3 MB total)
  float* qp    = (float*)d_ws;                                // 4*320*256
  float* kp    = qp + (size_t)B_ * QPAD_ * CDIM_;             // 4*256*10000
  float* stats = kp + (size_t)B_ * CDIM_ * HW_;               // 1200*2

  qproj_kernel<<<dim3(QPAD_ / 16, CDIM_ / 16, B_), 32, 0, stream>>>(q, q_w, q_b, qp);
  kproj_kernel<<<dim3(CDIM_ / 64, HW_ / 16, B_), 32, 0, stream>>>(k, k_w, k_b, kp);
  scores_kernel<<<dim3(QPAD_ / 64, HW_ / 16, B_ * NH_), 32, 0, stream>>>(qp, kp, mask, out);
  rowstat_kernel<<<dim3(B_ * Q_), 256, 0, stream>>>(out, stats);
  norm_kernel<<<dim3(93750), 256, 0, stream>>>(out, stats);
}